// BatchSuperQMulti_24120536334612
// MI455X (gfx1250) — compile-verified
//
#include <hip/hip_runtime.h>
#include <math.h>

// Problem constants (match reference)
#define B_SZ   8
#define N_PRIM 48
#define M_PTS  65536
#define MIN_S_F 0.001f
#define EPS_F   1e-6f
#define PCLAMP  1e6f

// Tiling: 8 waves/block, each wave owns 2 groups of 16 points -> 256 pts/block
#define WAVES_PER_BLOCK 8
#define THREADS        (WAVES_PER_BLOCK * 32)
#define PTS_PER_WAVE    32
#define PTS_PER_BLOCK  (WAVES_PER_BLOCK * PTS_PER_WAVE)     // 256
#define BLOCKS_PER_BATCH (M_PTS / PTS_PER_BLOCK)            // 256

// Per-primitive LDS record:
// [0..2]=R[0][i] [3..5]=R[1][i] [6..8]=R[2][i]  (A[i][j] = R[j][i])
// [9..11]=ofs_i = (R^T t)_i
// [12..14]=1/sx,1/sy,1/sz  [15]=1/e1 [16]=1/e2 [17]=e2/e1 [18]=-e1/2 [19]=exist
#define PSTRIDE 20

typedef __attribute__((ext_vector_type(2))) float v2f;
typedef __attribute__((ext_vector_type(8))) float v8f;

__device__ __forceinline__ float clampP(float v) {
    return fminf(fmaxf(fabsf(v), EPS_F), PCLAMP);
}
// pow for already-clamped positive base: exp2(p * log2(x)) on the trans pipe
__device__ __forceinline__ float powp(float xc, float p) {
    return __builtin_amdgcn_exp2f(p * __builtin_amdgcn_logf(xc));
}
__device__ __forceinline__ float sgn_clamp(float v) {
    return (v > 0.f ? 1.f : -1.f) * fmaxf(fabsf(v), EPS_F);
}
__device__ __forceinline__ float sigmoidf(float x) {
    return 1.f / (1.f + __builtin_amdgcn_exp2f(-x * 1.4426950408889634f));
}

// One SDF evaluation given rotated/translated coords (x,y,z) and param ptr
__device__ __forceinline__ float sq_sdf(float x, float y, float z,
                                        const float* __restrict__ pq) {
    float xs = sgn_clamp(x), ys = sgn_clamp(y), zs = sgn_clamp(z);
    float r0 = __builtin_amdgcn_sqrtf(xs*xs + ys*ys + zs*zs);
    float u1 = clampP(xs * pq[12]);
    float u2 = clampP(ys * pq[13]);
    float u3 = clampP(zs * pq[14]);
    float t1 = powp(clampP(u1*u1), pq[16]);     // ^(1/e2)
    float t2 = powp(clampP(u2*u2), pq[16]);     // ^(1/e2)
    float t3 = powp(clampP(u3*u3), pq[15]);     // ^(1/e1)
    float g  = powp(clampP(t1 + t2), pq[17]) + t3;   // ^(e2/e1)
    float f  = powp(clampP(g), pq[18]);              // ^(-e1/2)
    float sdf = r0 * (1.f - f);
    sdf = fminf(fmaxf(sdf, -PCLAMP), PCLAMP);        // safe_mul clamp
    return (pq[19] != 0.f) ? sdf : __builtin_inff(); // exist mask
}

__global__ __launch_bounds__(THREADS)
void sq_sdf_kernel(const float* __restrict__ points,
                   const float* __restrict__ raw_scale,
                   const float* __restrict__ raw_exp,
                   const float* __restrict__ raw_rot,
                   const float* __restrict__ trans,
                   const int*   __restrict__ exist,
                   float* __restrict__ out) {
    __shared__ float sp[N_PRIM * PSTRIDE];

    const int tid   = threadIdx.x;
    const int b     = blockIdx.x / BLOCKS_PER_BATCH;
    const int chunk = blockIdx.x % BLOCKS_PER_BATCH;

    // ---- per-primitive parameter precompute (one thread per primitive) ----
    if (tid < N_PRIM) {
        const int gi = b * N_PRIM + tid;
        const float* q = raw_rot + gi * 4;
        float qw = q[0], qx = q[1], qy = q[2], qz = q[3];
        float inv = rsqrtf(qw*qw + qx*qx + qy*qy + qz*qz);
        qw *= inv; qx *= inv; qy *= inv; qz *= inv;
        float r00 = 1.f - 2.f*(qy*qy + qz*qz);
        float r01 = 2.f*(qx*qy - qw*qz);
        float r02 = 2.f*(qx*qz + qw*qy);
        float r10 = 2.f*(qx*qy + qw*qz);
        float r11 = 1.f - 2.f*(qx*qx + qz*qz);
        float r12 = 2.f*(qy*qz - qw*qx);
        float r20 = 2.f*(qx*qz - qw*qy);
        float r21 = 2.f*(qy*qz + qw*qx);
        float r22 = 1.f - 2.f*(qx*qx + qy*qy);
        float* pp = sp + tid * PSTRIDE;
        pp[0] = r00; pp[1] = r01; pp[2] = r02;
        pp[3] = r10; pp[4] = r11; pp[5] = r12;
        pp[6] = r20; pp[7] = r21; pp[8] = r22;
        const float* tr = trans + gi * 3;
        float t0 = tr[0], t1 = tr[1], t2 = tr[2];
        // ofs_i = sum_j R[j][i] * t_j   (X = R^T p - ofs)
        pp[9]  = r00*t0 + r10*t1 + r20*t2;
        pp[10] = r01*t0 + r11*t1 + r21*t2;
        pp[11] = r02*t0 + r12*t1 + r22*t2;
        const float* rs = raw_scale + gi * 3;
        pp[12] = 1.f / (expf(rs[0]) + MIN_S_F);
        pp[13] = 1.f / (expf(rs[1]) + MIN_S_F);
        pp[14] = 1.f / (expf(rs[2]) + MIN_S_F);
        const float* re = raw_exp + gi * 2;
        float e1 = sigmoidf(re[0]) * 1.8f + 0.1f;
        float e2 = sigmoidf(re[1]) * 1.8f + 0.1f;
        pp[15] = 1.f / e1;
        pp[16] = 1.f / e2;
        pp[17] = e2 / e1;
        pp[18] = -0.5f * e1;
        pp[19] = exist[gi] ? 1.f : 0.f;
    }
    __syncthreads();

    const int  lane = tid & 31;
    const int  wave = tid >> 5;
    const bool hiHalf = lane >= 16;
    const int  col  = lane & 15;

    // Two 16-point groups per wave
    const int pt0 = chunk * PTS_PER_BLOCK + wave * PTS_PER_WAVE + col;
    const int pt1 = pt0 + 16;
    const float* P = points + b * 3 * M_PTS;

    // B fragment (K x N = 4 x 16 f32): lanes 0-15 hold rows K=0,1; lanes 16-31
    // hold rows K=2,3. Row 3 == 1.0 carries the translation fold.
    v2f bf0, bf1;
    if (!hiHalf) {
        bf0.x = P[pt0];            bf0.y = P[M_PTS + pt0];
        bf1.x = P[pt1];            bf1.y = P[M_PTS + pt1];
    } else {
        bf0.x = P[2*M_PTS + pt0];  bf0.y = 1.0f;
        bf1.x = P[2*M_PTS + pt1];  bf1.y = 1.0f;
    }

    // A fragment indexing: lane's row M = col; rows 0-2 -> primitive 2*pair,
    // rows 8-10 -> primitive 2*pair+1; lanes 0-15 supply K=0,1; 16-31 K=2,3.
    const int  mp     = col & 7;
    const int  rowSel = col >> 3;        // which primitive this A-row belongs to
    const bool validRow = (mp < 3);

    float mn0 = __builtin_inff();
    float mn1 = __builtin_inff();

    #pragma unroll 2
    for (int pair = 0; pair < N_PRIM / 2; ++pair) {
        v2f a; a.x = 0.f; a.y = 0.f;
        if (validRow) {
            const float* pa = sp + (pair * 2 + rowSel) * PSTRIDE;
            if (!hiHalf) { a.x = pa[mp];     a.y = pa[3 + mp]; }   // R[0][m], R[1][m]
            else         { a.x = pa[6 + mp]; a.y = -pa[9 + mp]; }  // R[2][m], -ofs[m]
        }
        v8f c0 = {0.f,0.f,0.f,0.f,0.f,0.f,0.f,0.f};
        v8f c1 = {0.f,0.f,0.f,0.f,0.f,0.f,0.f,0.f};
        c0 = __builtin_amdgcn_wmma_f32_16x16x4_f32(false, a, false, bf0,
                                                   (short)0, c0, false, false);
        c1 = __builtin_amdgcn_wmma_f32_16x16x4_f32(false, a, false, bf1,
                                                   (short)0, c1, false, false);
        // C layout: VGPR v, lanes 0-15 -> row v (prim n0); lanes 16-31 -> row v+8 (n1)
        const float* pq = sp + (pair * 2 + (hiHalf ? 1 : 0)) * PSTRIDE;
        mn0 = fminf(mn0, sq_sdf(c0[0], c0[1], c0[2], pq));
        mn1 = fminf(mn1, sq_sdf(c1[0], c1[1], c1[2], pq));
    }

    // lanes l / l+16 hold mins over even / odd primitives of the same point
    mn0 = fminf(mn0, __shfl_xor(mn0, 16, 32));
    mn1 = fminf(mn1, __shfl_xor(mn1, 16, 32));
    if (!hiHalf) {
        out[b * M_PTS + pt0] = mn0;
        out[b * M_PTS + pt1] = mn1;
    }
}

extern "C" void kernel_launch(void* const* d_in, const int* in_sizes, int n_in,
                              void* d_out, int out_size, void* d_ws, size_t ws_size,
                              hipStream_t stream) {
    const float* points    = (const float*)d_in[0];
    const float* raw_scale = (const float*)d_in[1];
    const float* raw_exp   = (const float*)d_in[2];
    const float* raw_rot   = (const float*)d_in[3];
    const float* trans     = (const float*)d_in[4];
    const int*   exist     = (const int*)d_in[5];
    float* out = (float*)d_out;

    dim3 grid(B_SZ * BLOCKS_PER_BATCH);   // 8 * 256 = 2048 blocks
    sq_sdf_kernel<<<grid, THREADS, 0, stream>>>(points, raw_scale, raw_exp,
                                                raw_rot, trans, exist, out);
}